// Torch_ROI_16106127360674
// MI455X (gfx1250) — compile-verified
//
#include <hip/hip_runtime.h>

typedef __attribute__((ext_vector_type(2))) float v2f;
typedef __attribute__((ext_vector_type(8))) float v8f;

#define N_ROI 512
#define C_CH  1024
#define HW    14
#define FS    14
#define ROI_SCALE (1.0f / 16.0f)
#define LSTRIDE 18           // padded LDS row stride (dwords): conflict-free halves
#define TILE   (HW * HW)     // 196

// ---------------------------------------------------------------------------
// Prep kernel: per ROI, build zero-padded 16x16 row-pool (Mr) and col-pool (Mc)
// matrices plus the batch index. One thread per (roi, output-bin) pair.
// Exactly mirrors the reference _pool_matrix integer bin math.
// ---------------------------------------------------------------------------
__global__ __launch_bounds__(256) void roi_prep(const float* __restrict__ ROI,
                                                float* __restrict__ Mr,
                                                float* __restrict__ Mc,
                                                int* __restrict__ fmno) {
  int t = blockIdx.x * blockDim.x + threadIdx.x;
  if (t >= N_ROI * 16) return;
  int n = t >> 4;
  int i = t & 15;
  const float* r = ROI + n * 5;
  if (i == 0) fmno[n] = (int)r[0];

  float x1f = floorf(r[1] * ROI_SCALE);  // x -> rows (H)
  float y1f = floorf(r[2] * ROI_SCALE);  // y -> cols (W)
  float x2f = ceilf(r[3] * ROI_SCALE);
  float y2f = ceilf(r[4] * ROI_SCALE);
  int x1 = (int)x1f; x1 = x1 > 0 ? x1 : 0;
  int y1 = (int)y1f; y1 = y1 > 0 ? y1 : 0;
  int x2 = (int)x2f; x2 = x2 < FS ? x2 : FS;
  int y2 = (int)y2f; y2 = y2 < FS ? y2 : FS;
  int Lx = x2 - x1; Lx = Lx > 1 ? Lx : 1;
  int Ly = y2 - y1; Ly = Ly > 1 ? Ly : 1;

  float* mrow = Mr + n * 256 + i * 16;
  float* mcol = Mc + n * 256 + i * 16;

  if (i >= FS) {  // padding rows of the 16x16 tiles
    for (int p = 0; p < 16; ++p) { mrow[p] = 0.0f; mcol[p] = 0.0f; }
    return;
  }
  {
    int b0 = (i * Lx) / FS;
    int b1 = ((i + 1) * Lx + (FS - 1)) / FS;
    int cnt = b1 - b0; cnt = cnt > 1 ? cnt : 1;
    float inv = 1.0f / (float)cnt;
    for (int p = 0; p < 16; ++p) {
      int rel = p - x1;
      mrow[p] = (p < FS && rel >= b0 && rel < b1) ? inv : 0.0f;
    }
  }
  {
    int b0 = (i * Ly) / FS;
    int b1 = ((i + 1) * Ly + (FS - 1)) / FS;
    int cnt = b1 - b0; cnt = cnt > 1 ? cnt : 1;
    float inv = 1.0f / (float)cnt;
    for (int p = 0; p < 16; ++p) {
      int rel = p - y1;
      mcol[p] = (p < FS && rel >= b0 && rel < b1) ? inv : 0.0f;
    }
  }
}

// ---------------------------------------------------------------------------
// Main kernel: one block per ROI (8 wave32). Each wave strides over channels.
// Per channel (all I/O coalesced, single divergent store at the tail):
//   7x coalesced global loads -> padded LDS (zeroed halo, no predication)
//   A-layout LDS reads of F;  V = F * Mc^T   (4x wmma_f32_16x16x4, K=16)
//   D-layout -> LDS -> B-layout;  D = Mr * V (4x wmma_f32_16x16x4)
//   D-layout -> LDS -> compact -> 7x coalesced global stores
// Layouts (f32 WMMA, wave32):
//   A (16x4):  vgpr v, lane l -> M = l&15,           K = v + 2*(l>>4)
//   B (4x16):  vgpr v, lane l -> K = v + 2*(l>>4),   N = l&15
//   C/D:       vgpr r, lane l -> M = r + 8*(l>>4),   N = l&15
// ---------------------------------------------------------------------------
__global__ __launch_bounds__(256) void roi_pool_wmma(
    const float* __restrict__ tensor, const float* __restrict__ Mr,
    const float* __restrict__ Mc, const int* __restrict__ fmno,
    float* __restrict__ out) {
  __shared__ float ldsF[8][16 * LSTRIDE];  // F staging, halo stays zero
  __shared__ float ldsV[8][16 * LSTRIDE];  // V / D bounce

  const int n    = blockIdx.x;
  const int lane = threadIdx.x & 31;
  const int wv   = threadIdx.x >> 5;
  const int half = lane >> 4;   // 0 or 1
  const int lj   = lane & 15;   // 0..15

  float* lf = &ldsF[wv][0];
  float* lv = &ldsV[wv][0];

  // Zero the F staging buffer once: padding positions stay zero forever,
  // so A-layout reads never need predication. 288 = 9 * 32.
#pragma unroll
  for (int j = 0; j < 9; ++j) lf[lane + j * 32] = 0.0f;

  // Per-lane compact<->padded index maps (loop-invariant, live in VGPRs).
  int gofs[7], gl[7], pidx[7];
#pragma unroll
  for (int t = 0; t < 7; ++t) {
    int g = t * 32 + lane;
    gofs[t] = g;
    int gc = g < (TILE - 1) ? g : (TILE - 1);  // clamp: tail lanes alias (13,13)
    gl[t] = gc;
    pidx[t] = (gc / FS) * LSTRIDE + (gc % FS);
  }
  const bool tail_ok = lane < (TILE - 6 * 32);  // 196-192 = 4 valid tail lanes

  // Constant operands for this ROI:
  //   MrA: A-layout chunks of Mr;  McB: B-layout chunks of Mc^T
  // B element (K=k, N=lj) of Mc^T is Mc[lj][k] -> same indexing expression.
  const float* mr = Mr + n * 256;
  const float* mc = Mc + n * 256;
  v2f MrA[4], McB[4];
#pragma unroll
  for (int kb = 0; kb < 4; ++kb) {
#pragma unroll
    for (int v = 0; v < 2; ++v) {
      int k = kb * 4 + v + 2 * half;
      MrA[kb][v] = mr[lj * 16 + k];
      McB[kb][v] = mc[lj * 16 + k];
    }
  }

  const int fm = fmno[n];
  const float* F = tensor + (size_t)fm * (size_t)(C_CH * TILE) + (size_t)wv * TILE;
  float* o = out + ((size_t)n * C_CH + wv) * TILE;
  const int dbase = half * (8 * LSTRIDE) + lj;  // D-layout LDS store base

  for (int it = 0; it < C_CH / 8; ++it) {
    if (it + 1 < C_CH / 8)
      __builtin_prefetch(F + 8 * TILE, 0, 1);  // next tile -> global_prefetch

    // ---- Stage F: coalesced, branch-free (tail lanes clamp to element 195
    //      and redundantly rewrite (13,13) with its own value).
#pragma unroll
    for (int t = 0; t < 7; ++t) {
      float val = F[gl[t]];
      lf[pidx[t]] = val;
    }

    // ---- Read F in A layout (consecutive K pairs -> ds_load_b64, no masks)
    v2f Fa[4];
#pragma unroll
    for (int kb = 0; kb < 4; ++kb) {
      int k0 = kb * 4 + 2 * half;
      Fa[kb][0] = lf[lj * LSTRIDE + k0];
      Fa[kb][1] = lf[lj * LSTRIDE + k0 + 1];
    }

    // ---- V = F * Mc^T  (accumulate K=16 as 4 chained K=4 WMMAs)
    v8f acc = {};
#pragma unroll
    for (int kb = 0; kb < 4; ++kb)
      acc = __builtin_amdgcn_wmma_f32_16x16x4_f32(false, Fa[kb], false, McB[kb],
                                                  (short)0, acc, false, false);

    // ---- V: D-layout -> LDS (per-wave region; DS in-order, no barriers)
#pragma unroll
    for (int rr = 0; rr < 8; ++rr)
      lv[dbase + rr * LSTRIDE] = acc[rr];

    // ---- Reload V in B layout
    v2f Vb[4];
#pragma unroll
    for (int kb = 0; kb < 4; ++kb) {
#pragma unroll
      for (int v = 0; v < 2; ++v) {
        int k = kb * 4 + v + 2 * half;
        Vb[kb][v] = lv[k * LSTRIDE + lj];
      }
    }

    // ---- D = Mr * V
    v8f d = {};
#pragma unroll
    for (int kb = 0; kb < 4; ++kb)
      d = __builtin_amdgcn_wmma_f32_16x16x4_f32(false, MrA[kb], false, Vb[kb],
                                                (short)0, d, false, false);

    // ---- D: D-layout -> LDS (reuse lv; prior Vb loads already issued)
#pragma unroll
    for (int rr = 0; rr < 8; ++rr)
      lv[dbase + rr * LSTRIDE] = d[rr];

    // ---- Compact read + fully coalesced contiguous stores (784 B/tile).
    //      Only the final partial store is predicated.
#pragma unroll
    for (int t = 0; t < 7; ++t) {
      float val = lv[pidx[t]];
      if (t < 6) {
        o[gofs[t]] = val;
      } else if (tail_ok) {
        o[gofs[t]] = val;
      }
    }

    F += 8 * TILE;
    o += 8 * TILE;
  }
}

extern "C" void kernel_launch(void* const* d_in, const int* in_sizes, int n_in,
                              void* d_out, int out_size, void* d_ws, size_t ws_size,
                              hipStream_t stream) {
  const float* tensor = (const float*)d_in[0];  // [8,1024,14,14] f32
  const float* ROI    = (const float*)d_in[1];  // [512,5] f32
  float* out = (float*)d_out;                   // [512,1024,14,14] f32

  // Workspace: Mr (512*256 f32) | Mc (512*256 f32) | fmno (512 i32)  ~1.05 MB
  float* Mr = (float*)d_ws;
  float* Mc = Mr + N_ROI * 256;
  int* fmno = (int*)(Mc + N_ROI * 256);

  roi_prep<<<(N_ROI * 16 + 255) / 256, 256, 0, stream>>>(ROI, Mr, Mc, fmno);
  roi_pool_wmma<<<N_ROI, 256, 0, stream>>>(tensor, Mr, Mc, fmno, out);
}